// SoftPatternClassifier_73710228734005
// MI455X (gfx1250) — compile-verified
//
#include <hip/hip_runtime.h>

typedef __attribute__((ext_vector_type(2))) float v2f;
typedef __attribute__((ext_vector_type(8))) float v8f;

#define EMB_DIM 300
#define N_LOCAL 10000
#define NPAT    200
#define LM1D    5
#define JDIM    (NPAT * LM1D)   // 1000
#define BATCH   128
#define TLEN    256
#define NTILES  5               // N tiles per wave: 625 = 5 * 125

// ---------------------------------------------------------------------------
// Kernel A: transition table via V_WMMA_F32_16X16X4_F32.
// C[j][v] = sum_k diags[j][k]*emb[k][v] + bias[j], stored scan-friendly as
//   table[v*1000 + (j%5)*200 + (j/5)]   (i.e. [token][state][pattern]).
// One wave computes a 16(M=j) x 80(N=v) strip: 5 accumulators, A reused 5x.
// M: 1000 -> 63 tiles (pad rows clamped, never stored). N: 10000 = 80*125.
// K: 300 = 4*75.
// ---------------------------------------------------------------------------
__global__ __launch_bounds__(32) void sopa_trans_wmma(
    const float* __restrict__ diags,   // (1000, 300) row-major
    const float* __restrict__ bias,    // (1000,)
    const float* __restrict__ emb,     // (300, 10000) row-major
    float* __restrict__ table)         // (10000, 5, 200)
{
    const int lane = threadIdx.x;      // 0..31
    const int lm   = lane & 15;        // M (for A) / N (for B,C,D)
    const int half = lane >> 4;        // K-half selector
    const int m0   = blockIdx.x * 16;          // j tile base
    const int n0   = blockIdx.y * (16 * NTILES); // v strip base

    // C/D layout: VGPR r, lanes 0-15 -> M = r ; lanes 16-31 -> M = r+8 ; N = lane%16
    v8f c[NTILES];
#pragma unroll
    for (int r = 0; r < 8; ++r) {
        int j = m0 + r + 8 * half;
        j = (j < JDIM) ? j : (JDIM - 1);       // clamp: pad rows discarded at store
        const float bj = bias[j];
#pragma unroll
        for (int tile = 0; tile < NTILES; ++tile) c[tile][r] = bj;
    }

    int mrow = m0 + lm;                        // A row (j)
    mrow = (mrow < JDIM) ? mrow : (JDIM - 1);  // clamp instead of predicating loads

    for (int k0 = 0; k0 < EMB_DIM; k0 += 4) {
        const int krow = k0 + half * 2;        // even -> 8B-aligned b64 load
        const v2f a = *(const v2f*)(diags + (size_t)mrow * EMB_DIM + krow);

        v2f b[NTILES];
#pragma unroll
        for (int tile = 0; tile < NTILES; ++tile) {
#pragma unroll
            for (int r = 0; r < 2; ++r)        // B 4x16: K = krow+r, N = n0+tile*16+lm
                b[tile][r] = emb[(size_t)(krow + r) * N_LOCAL + n0 + tile * 16 + lm];
        }
#pragma unroll
        for (int tile = 0; tile < NTILES; ++tile)
            c[tile] = __builtin_amdgcn_wmma_f32_16x16x4_f32(
                          false, a, false, b[tile], (short)0, c[tile], false, false);
    }

#pragma unroll
    for (int r = 0; r < 8; ++r) {
        const int j = m0 + r + 8 * half;
        if (j < JDIM) {
            const int pidx = j / 5, s = j % 5;
#pragma unroll
            for (int tile = 0; tile < NTILES; ++tile) {
                const int v = n0 + tile * 16 + lm;
                table[(size_t)v * JDIM + s * NPAT + pidx] = c[tile][r];
            }
        }
    }
}

// ---------------------------------------------------------------------------
// Kernel B: max-plus scan + epilogue. One block per document, one thread per
// pattern. Coalesced [token][state][pattern] loads, register double-buffer
// (t+1) and global_prefetch (t+2) to hide the serial L2 latency chain.
// (s-mu)/sqrt(var+eps) > 0  <=>  s > mu, so variance is never needed.
// ---------------------------------------------------------------------------
__global__ __launch_bounds__(256) void sopa_scan(
    const float* __restrict__ table,     // (10000, 5, 200)
    const float* __restrict__ wildcards, // (200, 5)
    const float* __restrict__ linear_w,  // (2, 200)
    const float* __restrict__ linear_b,  // (2,)
    const int*   __restrict__ docs,      // (128, 256)
    const int*   __restrict__ doc_lens,  // (128,)
    float* __restrict__ out)             // (128, 2)
{
    __shared__ int   s_tok[TLEN];
    __shared__ float s_val[NPAT];
    __shared__ float s_mu[1];

    const int b   = blockIdx.x;
    const int tid = threadIdx.x;
    const int p   = tid;
    const int pp  = (p < NPAT) ? p : 0;        // clamp for safe (discarded) lanes

    int len = doc_lens[b];
    if (len > TLEN) len = TLEN;
    if (len < 0)    len = 0;

    s_tok[tid] = docs[b * TLEN + tid];
    __syncthreads();

    const float w0 = wildcards[pp * LM1D + 0];
    const float w1 = wildcards[pp * LM1D + 1];
    const float w2 = wildcards[pp * LM1D + 2];
    const float w3 = wildcards[pp * LM1D + 3];
    const float w4 = wildcards[pp * LM1D + 4];
    const int   es = 5 - pp / 50;              // end state: 5,4,3,2 per 50-pattern group

    const float NEG = -__builtin_inff();
    float h1 = NEG, h2 = NEG, h3 = NEG, h4 = NEG, h5 = NEG;
    float sc = NEG;

    // software pipeline: c* hold step t's transitions, u* prefetch step t+1
    float c0 = 0, c1 = 0, c2 = 0, c3 = 0, c4 = 0;
    if (len > 0) {
        const float* r0 = table + (size_t)s_tok[0] * JDIM + pp;
        c0 = r0[0 * NPAT]; c1 = r0[1 * NPAT]; c2 = r0[2 * NPAT];
        c3 = r0[3 * NPAT]; c4 = r0[4 * NPAT];
    }

    for (int t = 0; t < len; ++t) {
        float u0 = 0, u1 = 0, u2 = 0, u3 = 0, u4 = 0;
        if (t + 1 < len) {                     // block-uniform branch
            const float* rn = table + (size_t)s_tok[t + 1] * JDIM + pp;
            u0 = rn[0 * NPAT]; u1 = rn[1 * NPAT]; u2 = rn[2 * NPAT];
            u3 = rn[3 * NPAT]; u4 = rn[4 * NPAT];
        }
        if (t + 2 < len)
            __builtin_prefetch(table + (size_t)s_tok[t + 2] * JDIM + pp, 0, 1);

        const float e0 = fmaxf(c0, w0);
        const float e1 = fmaxf(c1, w1);
        const float e2 = fmaxf(c2, w2);
        const float e3 = fmaxf(c3, w3);
        const float e4 = fmaxf(c4, w4);

        const float n5 = h4 + e4;
        const float n4 = h3 + e3;
        const float n3 = h2 + e2;
        const float n2 = h1 + e1;
        const float n1 = e0;                   // restart state contributes 0
        h5 = n5; h4 = n4; h3 = n3; h2 = n2; h1 = n1;

        const float endv = (es == 5) ? h5 : (es == 4) ? h4 : (es == 3) ? h3 : h2;
        sc = fmaxf(sc, endv);

        c0 = u0; c1 = u1; c2 = u2; c3 = u3; c4 = u4;
    }

    if (p < NPAT) s_val[p] = sc;
    __syncthreads();

    if (tid == 0) {
        // replace -inf with the finite row minimum, then take the mean
        float mn = __builtin_inff();
        for (int i = 0; i < NPAT; ++i) {
            const float v = s_val[i];
            if (v != NEG && v < mn) mn = v;
        }
        float sum = 0.0f;
        for (int i = 0; i < NPAT; ++i) {
            float v = s_val[i];
            if (v == NEG) { v = mn; s_val[i] = v; }
            sum += v;
        }
        s_mu[0] = sum / (float)NPAT;
    }
    __syncthreads();

    const float mu = s_mu[0];
    if (p < NPAT) s_val[p] = (s_val[p] > mu) ? 1.0f : 0.0f;
    __syncthreads();

    if (tid < 2) {   // deterministic 200-element dot per output class
        float acc = 0.0f;
        for (int i = 0; i < NPAT; ++i)
            acc += s_val[i] * linear_w[tid * NPAT + i];
        out[b * 2 + tid] = acc + linear_b[tid];
    }
}

// ---------------------------------------------------------------------------
// Launch
// ---------------------------------------------------------------------------
extern "C" void kernel_launch(void* const* d_in, const int* in_sizes, int n_in,
                              void* d_out, int out_size, void* d_ws, size_t ws_size,
                              hipStream_t stream) {
    const float* emb       = (const float*)d_in[0];  // (300, 10000)
    const float* diags     = (const float*)d_in[1];  // (1000, 300)
    const float* bias      = (const float*)d_in[2];  // (1000, 1)
    const float* wildcards = (const float*)d_in[3];  // (200, 5)
    const float* linear_w  = (const float*)d_in[4];  // (2, 200)
    const float* linear_b  = (const float*)d_in[5];  // (2,)
    const int*   docs      = (const int*)d_in[6];    // (128, 256)
    const int*   doc_lens  = (const int*)d_in[7];    // (128,)
    float* out   = (float*)d_out;
    float* table = (float*)d_ws;                     // 10000*1000*4 = 40 MB scratch

    dim3 gridA((JDIM + 15) / 16, N_LOCAL / (16 * NTILES));   // 63 x 125
    sopa_trans_wmma<<<gridA, 32, 0, stream>>>(diags, bias, emb, table);
    sopa_scan<<<BATCH, TLEN, 0, stream>>>(table, wildcards, linear_w, linear_b,
                                          docs, doc_lens, out);
}